// SkeletonModel32_55413668053388
// MI455X (gfx1250) — compile-verified
//
#include <hip/hip_runtime.h>
#include <hip/hip_bf16.h>

// ---------------------------------------------------------------------------
// FK skeleton kernel for MI455X (gfx1250, wave32).
// Bandwidth-bound (~186 MB @ 23.3 TB/s ~= 8us). Strategy:
//   * async global->LDS staging of each block's contiguous x chunk (b128)
//   * per-lane unrolled Rodrigues + FK chain in f32 VALU registers
//   * results staged in LDS, async LDS->global b128 stores
// ---------------------------------------------------------------------------

#define TPB 64                 // 2 wave32 waves per workgroup
#define XF  (TPB * 99)         // staged input floats per block
#define OF  (TPB * 78)         // staged output floats per block

#if defined(__gfx1250__) && __has_builtin(__builtin_amdgcn_global_load_async_to_lds_b128)
#define HAVE_ASYNC_LD 1
#endif
#if defined(__gfx1250__) && __has_builtin(__builtin_amdgcn_global_store_async_from_lds_b128)
#define HAVE_ASYNC_ST 1
#endif

#if __has_builtin(__builtin_amdgcn_s_wait_asynccnt)
#define WAIT_ASYNC0() __builtin_amdgcn_s_wait_asynccnt(0)
#else
#define WAIT_ASYNC0() asm volatile("s_wait_asynccnt 0" ::: "memory")
#endif

// Builtin param types (from hipcc diagnostic): pointer-to-v4i32, global (AS1)
// first, LDS (AS3) second.
typedef int v4i __attribute__((vector_size(16)));
typedef __attribute__((address_space(1))) v4i* gv4p;
typedef __attribute__((address_space(3))) v4i* lv4p;

#define DEV __device__ __forceinline__

struct M3 { float a00,a01,a02,a10,a11,a12,a20,a21,a22; };
struct V3 { float x,y,z; };

// Rodrigues, exactly mirroring the reference: k = v/max(|v|,1e-8),
// R = I + sin(t) K + (1-cos(t)) K^2
DEV M3 rodrigues(const float* v) {
  float vx = v[0], vy = v[1], vz = v[2];
  float d  = vx*vx + vy*vy + vz*vz;
  float t  = __builtin_sqrtf(d);
  float inv = __builtin_amdgcn_rcpf(fmaxf(t, 1e-8f));
  float kx = vx*inv, ky = vy*inv, kz = vz*inv;
  float s = __sinf(t), c = __cosf(t);
  float u = 1.0f - c;
  float kxky = kx*ky, kxkz = kx*kz, kykz = ky*kz;
  M3 r;
  r.a00 = 1.0f - u*(ky*ky + kz*kz);
  r.a01 = -s*kz + u*kxky;
  r.a02 =  s*ky + u*kxkz;
  r.a10 =  s*kz + u*kxky;
  r.a11 = 1.0f - u*(kx*kx + kz*kz);
  r.a12 = -s*kx + u*kykz;
  r.a20 = -s*ky + u*kxkz;
  r.a21 =  s*kx + u*kykz;
  r.a22 = 1.0f - u*(kx*kx + ky*ky);
  return r;
}

DEV M3 mul(const M3& A, const M3& B) {   // A @ B
  M3 r;
  r.a00 = A.a00*B.a00 + A.a01*B.a10 + A.a02*B.a20;
  r.a01 = A.a00*B.a01 + A.a01*B.a11 + A.a02*B.a21;
  r.a02 = A.a00*B.a02 + A.a01*B.a12 + A.a02*B.a22;
  r.a10 = A.a10*B.a00 + A.a11*B.a10 + A.a12*B.a20;
  r.a11 = A.a10*B.a01 + A.a11*B.a11 + A.a12*B.a21;
  r.a12 = A.a10*B.a02 + A.a11*B.a12 + A.a12*B.a22;
  r.a20 = A.a20*B.a00 + A.a21*B.a10 + A.a22*B.a20;
  r.a21 = A.a20*B.a01 + A.a21*B.a11 + A.a22*B.a21;
  r.a22 = A.a20*B.a02 + A.a21*B.a12 + A.a22*B.a22;
  return r;
}

DEV V3 rT_off(const M3& R, const float* o) {  // einsum('bji,j->bi'): R^T @ o
  V3 p;
  p.x = R.a00*o[0] + R.a10*o[1] + R.a20*o[2];
  p.y = R.a01*o[0] + R.a11*o[1] + R.a21*o[2];
  p.z = R.a02*o[0] + R.a12*o[1] + R.a22*o[2];
  return p;
}
DEV V3 addv(V3 a, V3 b) { return V3{a.x+b.x, a.y+b.y, a.z+b.z}; }

// Full per-pose FK. xr: 99 floats, off: 96 floats (uniform -> SMEM),
// outr: 78 floats. Pointers may be LDS or global; inliner + address-space
// inference picks ds_* vs global_* as appropriate.
DEV void compute_pose(const float* __restrict__ xr,
                      const float* __restrict__ off,
                      float* __restrict__ outr) {
  auto ST = [&](int slot, V3 p) {
    outr[slot*3+0] = p.x; outr[slot*3+1] = p.y; outr[slot*3+2] = p.z;
  };

  M3 Rhip = rodrigues(xr + 4);
  V3 pHip = { off[0] + xr[0], off[1] + xr[1], off[2] + xr[2] };
  ST(0, pHip);

  // right leg: rHip(1) rKnee(2) rAnkle(3) rToe(4)  [rToe rot never used]
  { M3 r = Rhip; V3 p = pHip;
    p = addv(rT_off(r, off+3),  p); ST(1, p); r = mul(rodrigues(xr+7),  r);
    p = addv(rT_off(r, off+6),  p); ST(2, p); r = mul(rodrigues(xr+10), r);
    p = addv(rT_off(r, off+9),  p); ST(3, p); r = mul(rodrigues(xr+13), r);
    p = addv(rT_off(r, off+12), p); ST(4, p);
  }
  // left leg: lHip(6) lKnee(7) lAnkle(8) lToe(9)
  { M3 r = Rhip; V3 p = pHip;
    p = addv(rT_off(r, off+18), p); ST(5, p); r = mul(rodrigues(xr+19), r);
    p = addv(rT_off(r, off+21), p); ST(6, p); r = mul(rodrigues(xr+22), r);
    p = addv(rT_off(r, off+24), p); ST(7, p); r = mul(rodrigues(xr+25), r);
    p = addv(rT_off(r, off+27), p); ST(8, p);
  }
  // spine chain: spine(11) spine1(12) thorax(13) neck(14) head(15)
  M3 rS1; V3 pS1;
  { M3 r = Rhip; V3 p = pHip;
    p = addv(rT_off(r, off+33), p); ST(9,  p); r = mul(rodrigues(xr+31), r);
    p = addv(rT_off(r, off+36), p); ST(10, p); r = mul(rodrigues(xr+34), r);
    rS1 = r; pS1 = p;
    p = addv(rT_off(r, off+39), p); ST(11, p); r = mul(rodrigues(xr+37), r);
    p = addv(rT_off(r, off+42), p); ST(12, p); r = mul(rodrigues(xr+40), r);
    p = addv(rT_off(r, off+45), p); ST(13, p);         // head: identity ang, leaf
  }
  // left arm from spine1: lShoulderAnchor(16) lShoulder(17) lElbow(18) lWrist(19)
  { M3 r = rS1; V3 p = pS1;
    p = addv(rT_off(r, off+48), p); ST(14, p); r = mul(rodrigues(xr+43), r);
    p = addv(rT_off(r, off+51), p); ST(15, p); r = mul(rodrigues(xr+46), r);
    p = addv(rT_off(r, off+54), p); ST(16, p); r = mul(rodrigues(xr+49), r);
    p = addv(rT_off(r, off+57), p); ST(17, p); r = mul(rodrigues(xr+52), r);
    V3 q = addv(rT_off(r, off+60), p); ST(18, q);      // lThumb(20)
    V3 w = addv(rT_off(r, off+66), p); ST(19, w);      // lWristEnd(22)
  }
  // right arm from spine1: rShoulderAnchor(24) rShoulder(25) rElbow(26) rWrist(27)
  { M3 r = rS1; V3 p = pS1;
    p = addv(rT_off(r, off+72), p); ST(20, p); r = mul(rodrigues(xr+61), r);
    p = addv(rT_off(r, off+75), p); ST(21, p); r = mul(rodrigues(xr+64), r);
    p = addv(rT_off(r, off+78), p); ST(22, p); r = mul(rodrigues(xr+67), r);
    p = addv(rT_off(r, off+81), p); ST(23, p); r = mul(rodrigues(xr+70), r);
    V3 q = addv(rT_off(r, off+84), p); ST(24, q);      // rThumb(28)
    V3 w = addv(rT_off(r, off+90), p); ST(25, w);      // rWristEnd(30)
  }
}

__global__ __launch_bounds__(TPB) void fk_skeleton_kernel(
    const float* __restrict__ x, const float* __restrict__ off,
    float* __restrict__ out, int n) {
  __shared__ __align__(16) float sx[XF];
  __shared__ __align__(16) float so[OF];

  const int  ltid = threadIdx.x;
  const long long base = (long long)blockIdx.x * TPB;

  if (base + TPB <= n) {
    // ---- stage x chunk (25,344 B, 16B-aligned, contiguous) into LDS ----
    const float* gx = x + base * 99;
#if HAVE_ASYNC_LD
    for (int i = ltid; i < XF / 4; i += TPB) {
      __builtin_amdgcn_global_load_async_to_lds_b128(
          (gv4p)(gx + i * 4), (lv4p)(&sx[i * 4]), 0, 0);
    }
    WAIT_ASYNC0();          // own wave's async loads complete
#else
    for (int i = ltid; i < XF / 4; i += TPB)
      ((float4*)sx)[i] = ((const float4*)gx)[i];
#endif
    __syncthreads();        // other waves' staged data visible

    // ---- per-lane FK (x row stride 99 dwords: LDS-bank conflict free) ----
    compute_pose(&sx[ltid * 99], off, &so[ltid * 78]);

    __syncthreads();        // all result rows in LDS

    // ---- stream results out as contiguous b128 ----
    float* go = out + base * 78;
#if HAVE_ASYNC_ST
    for (int i = ltid; i < OF / 4; i += TPB) {
      __builtin_amdgcn_global_store_async_from_lds_b128(
          (gv4p)(go + i * 4), (lv4p)(&so[i * 4]), 0, 0);
    }
    WAIT_ASYNC0();
#else
    for (int i = ltid; i < OF / 4; i += TPB)
      ((float4*)go)[i] = ((const float4*)so)[i];
#endif
  } else {
    // tail block (not hit for B=262144): direct per-thread path
    long long p = base + ltid;
    if (p < n) compute_pose(x + p * 99, off, out + p * 78);
  }
}

extern "C" void kernel_launch(void* const* d_in, const int* in_sizes, int n_in,
                              void* d_out, int out_size, void* d_ws, size_t ws_size,
                              hipStream_t stream) {
  const float* x   = (const float*)d_in[0];   // (B, 99) f32
  const float* off = (const float*)d_in[1];   // (32, 3) f32
  float*       out = (float*)d_out;           // (B, 78) f32
  const int n = in_sizes[0] / 99;
  const int blocks = (n + TPB - 1) / TPB;
  fk_skeleton_kernel<<<blocks, TPB, 0, stream>>>(x, off, out, n);
}